// LinearAttention_38233798869626
// MI455X (gfx1250) — compile-verified
//
#include <hip/hip_runtime.h>
#include <math.h>

// LinearAttention on MI455X (gfx1250), fp32 WMMA path + TDM double-buffered
// streaming in the kv kernel.
// x:(32,256,56,56) f32 | qk_w:(512,128) | qk_b:(512) | pe_w:(256,9) | pe_b:(256)
// ws layout (floats): q[32*256*3136] | k[32*256*3136] | kv[32*8*32*32] | kmean[32*8*32]

#define NPIX   3136   // 56*56
#define CDIM   256
#define NHEADS 8
#define HD     32

typedef float v2f __attribute__((ext_vector_type(2)));
typedef float v8f __attribute__((ext_vector_type(8)));
typedef unsigned int u32x4 __attribute__((ext_vector_type(4)));
typedef int i32x4 __attribute__((ext_vector_type(4)));
typedef int i32x8 __attribute__((ext_vector_type(8)));

static __device__ __forceinline__ v8f wmma4(v2f a, v2f b, v8f c) {
  // D = A(16x4 f32) * B(4x16 f32) + C(16x16 f32)
  return __builtin_amdgcn_wmma_f32_16x16x4_f32(false, a, false, b, (short)0, c,
                                               false, false);
}

static __device__ __forceinline__ float elu1(float v) {
  return v > 0.0f ? v + 1.0f : __expf(v);   // elu(v)+1
}

static __device__ __forceinline__ v2f lds2(const float* p) {
  return *(const v2f*)p;                    // 8B-aligned -> ds_load_b64
}

// TDM: 2D tile (tile_d rows of tile_n dwords, global row stride = stride_elems
// dwords) from global memory into LDS at lds_off, LDS rows padded to 68 dwords
// via D# pad fields (interval = 64 DW -> code 5, amount = 4 DW -> code 3).
static __device__ __forceinline__ void tdm_load_tile(unsigned lds_off,
                                                     const float* gptr,
                                                     unsigned tile_n,
                                                     unsigned tile_d,
                                                     unsigned stride_elems) {
  unsigned long long ga = (unsigned long long)(const void*)gptr;
  u32x4 g0;
  g0[0] = 1u;                                       // count=1 (valid user D#)
  g0[1] = lds_off;                                  // lds_addr (bytes)
  g0[2] = (unsigned)(ga & 0xffffffffu);             // global_addr[31:0]
  g0[3] = (unsigned)((ga >> 32) & 0x1ffffffu)       // global_addr[56:32]
          | (2u << 30);                             // type=2 ("image")
  i32x8 g1;
  g1[0] = (int)((2u << 16)                          // data_size = 4B
                | (1u << 20)                        // pad_enable
                | (5u << 22)                        // pad_interval: 64 dwords
                | (3u << 25));                      // pad_amount: 4 dwords
  g1[1] = (int)((tile_n & 0xffffu) << 16);          // tensor_dim0[15:0]
  g1[2] = (int)(((tile_n >> 16) & 0xffffu)          // tensor_dim0[31:16]
                | ((tile_d & 0xffffu) << 16));      // tensor_dim1[15:0]
  g1[3] = (int)(((tile_d >> 16) & 0xffffu)          // tensor_dim1[31:16]
                | ((tile_n & 0xffffu) << 16));      // tile_dim0
  g1[4] = (int)(tile_d & 0xffffu);                  // tile_dim1 (tile_dim2=0)
  g1[5] = (int)stride_elems;                        // tensor_dim0_stride[31:0]
  g1[6] = 0;
  g1[7] = 0;
  i32x4 z4 = {0, 0, 0, 0};                          // groups 2/3 unused (2D)
  i32x8 z8 = {0, 0, 0, 0, 0, 0, 0, 0};
  __builtin_amdgcn_tensor_load_to_lds(g0, g1, z4, z4, z8, 0);
}

// ---------------------------------------------------------------------------
// Kernel 1: qk = elu(conv1x1_grouped(x, W) + b) + 1   (two GEMMs M=256,K=128)
// grid (49 n-tiles, 4 m-tiles of 128, 32 batches), block 256 = 8 waves (4Mx2N)
// All staged tiles K-innermost -> every WMMA fragment is one ds_load_b64.
// ---------------------------------------------------------------------------
__global__ __launch_bounds__(256)
void qk_gemm_kernel(const float* __restrict__ x, const float* __restrict__ w,
                    const float* __restrict__ bias, float* __restrict__ qbuf,
                    float* __restrict__ kbuf) {
  __shared__ float As[128][20];  // [m][k], stride 20: conflict-free b64 reads
  __shared__ float Bs[64][20];   // [n][k] (transposed at store time)
  const int t = threadIdx.x;
  const int lane = t & 31, wave = t >> 5;
  const int half = lane >> 4, lm = lane & 15;
  const int mw = (wave & 3) * 32, nw = (wave >> 2) * 32;
  const int n0 = blockIdx.x * 64;
  const int c0 = blockIdx.y * 128;          // output-channel base (0..511)
  const int cb = (c0 >= 256) ? 128 : 0;     // input-channel base per group
  const int b  = blockIdx.z;
  const float* xb = x + ((long)b * CDIM + cb) * NPIX + n0;

  v8f acc[2][2] = {};
  for (int k0 = 0; k0 < 128; k0 += 16) {
#pragma unroll
    for (int i = 0; i < 8; ++i) {           // stage A: 128m x 16k (K-innermost)
      int lin = t + i * 256;
      int k = lin & 15, m = lin >> 4;
      As[m][k] = w[(c0 + m) * 128 + k0 + k];
    }
#pragma unroll
    for (int i = 0; i < 4; ++i) {           // stage B: global coalesced in n
      int lin = t + i * 256;
      int n = lin & 63, k = lin >> 6;
      Bs[n][k] = xb[(long)(k0 + k) * NPIX + n];
    }
    __syncthreads();
#pragma unroll
    for (int kk = 0; kk < 16; kk += 4) {
      const int kc = kk + 2 * half;         // even -> 8B aligned
      v2f a0 = lds2(&As[mw + lm][kc]);
      v2f a1 = lds2(&As[mw + 16 + lm][kc]);
      v2f b0 = lds2(&Bs[nw + lm][kc]);
      v2f b1 = lds2(&Bs[nw + 16 + lm][kc]);
      acc[0][0] = wmma4(a0, b0, acc[0][0]);
      acc[0][1] = wmma4(a0, b1, acc[0][1]);
      acc[1][0] = wmma4(a1, b0, acc[1][0]);
      acc[1][1] = wmma4(a1, b1, acc[1][1]);
    }
    __syncthreads();
  }

  float* dst = (c0 < 256) ? qbuf : kbuf;    // uniform per block
  const int crel0 = (c0 < 256) ? c0 : (c0 - 256);
#pragma unroll
  for (int mi = 0; mi < 2; ++mi)
#pragma unroll
    for (int ni = 0; ni < 2; ++ni)
#pragma unroll
      for (int r = 0; r < 8; ++r) {
        int m_local = mw + mi * 16 + r + 8 * half;
        int n_local = nw + ni * 16 + lm;
        float v = acc[mi][ni][r] + bias[c0 + m_local];
        dst[((long)b * CDIM + crel0 + m_local) * NPIX + n0 + n_local] = elu1(v);
      }
}

// ---------------------------------------------------------------------------
// Kernel 2: kv[b,h,d,e] = s^2 * sum_n k[d,n]*v[e,n];  kmean[b,h,d] = mean_n k
// grid (8 heads, 32 batches), block 128 = 4 waves = 2x2 quadrants of 32x32.
// TDM double-buffered streaming: wave 0 issues tensor_load_to_lds for tile
// i+1, s_wait_tensorcnt(2) guarantees tile i landed (in-order completion),
// then all waves compute out of the padded (stride-68) LDS tiles.
// ---------------------------------------------------------------------------
__global__ __launch_bounds__(128)
void kv_kernel(const float* __restrict__ x, const float* __restrict__ kbuf,
               float* __restrict__ kvbuf, float* __restrict__ kmean) {
  __shared__ float kS[2][32][68];   // [buf][d][n], TDM-padded rows
  __shared__ float vS[2][32][68];
  __shared__ float ksum[32];
  const int t = threadIdx.x;
  const int lane = t & 31, wave = t >> 5;
  const int half = lane >> 4, lm = lane & 15;
  const int dq = (wave >> 1) * 16, eq = (wave & 1) * 16;
  const int h = blockIdx.x, b = blockIdx.y;
  const long base = ((long)b * CDIM + h * HD) * NPIX;  // k/v channel base

  if (t < 32) ksum[t] = 0.0f;

  if (wave == 0) {                          // prime the pipeline: tile 0
    tdm_load_tile((unsigned)(size_t)(void*)&kS[0][0][0], kbuf + base, 64, HD, NPIX);
    tdm_load_tile((unsigned)(size_t)(void*)&vS[0][0][0], x + base, 64, HD, NPIX);
  }

  float kacc = 0.0f;
  v8f acc = {};
  for (int it = 0; it < 49; ++it) {
    const int cur = it & 1;
    if (wave == 0) {
      if (it + 1 < 49) {                    // issue next tile, then drain to 2
        const long nx = base + (long)(it + 1) * 64;
        tdm_load_tile((unsigned)(size_t)(void*)&kS[cur ^ 1][0][0], kbuf + nx, 64, HD, NPIX);
        tdm_load_tile((unsigned)(size_t)(void*)&vS[cur ^ 1][0][0], x + nx, 64, HD, NPIX);
        __builtin_amdgcn_s_wait_tensorcnt(2);
      } else {
        __builtin_amdgcn_s_wait_tensorcnt(0);
      }
    }
    __syncthreads();                        // tile `cur` visible to all waves
#pragma unroll
    for (int kk = 0; kk < 64; kk += 4) {
      const int kc = kk + 2 * half;
      v2f a  = lds2(&kS[cur][dq + lm][kc]);
      v2f bb = lds2(&vS[cur][eq + lm][kc]);
      acc = wmma4(a, bb, acc);
    }
    {                                       // k row-sum partials from LDS
      const int d = t >> 2, c4 = (t & 3);
#pragma unroll
      for (int j = 0; j < 16; ++j) kacc += kS[cur][d][c4 + 4 * j];
    }
    __syncthreads();                        // done reading `cur` before reuse
  }

  atomicAdd(&ksum[t >> 2], kacc);           // ds_add_f32

  const float s2 = 1.0f / (float)NPIX;
  const long kvb = (long)(b * NHEADS + h) * HD * HD;
#pragma unroll
  for (int r = 0; r < 8; ++r) {
    int d = dq + r + 8 * half;
    int e = eq + lm;
    kvbuf[kvb + d * HD + e] = acc[r] * s2;
  }
  __syncthreads();
  if (t < 32) kmean[(long)(b * NHEADS + h) * HD + t] = ksum[t] * s2;
}

// ---------------------------------------------------------------------------
// Kernel 3: num = q^T(64n x 32d) @ kv(32d x 32e); out = num / (q.kmean + eps)
// grid (49 n-tiles, 8 heads, 32 batches), block 256 = 8 waves (4n x 2e tiles)
// ---------------------------------------------------------------------------
__global__ __launch_bounds__(256)
void out_kernel(const float* __restrict__ qbuf, const float* __restrict__ kvbuf,
                const float* __restrict__ kmean, float* __restrict__ out) {
  __shared__ float qS[64][34];    // [n][d], stride 34: conflict-free b64 reads
  __shared__ float kvT[32][34];   // [e][d]
  __shared__ float kms[32];
  __shared__ float dens[64];
  __shared__ float otile[32][65]; // [e][n] padded -> conflict-free scatter
  const int t = threadIdx.x;
  const int lane = t & 31, wave = t >> 5;
  const int half = lane >> 4, lm = lane & 15;
  const int m_off = (wave & 3) * 16, e_off = (wave >> 2) * 16;
  const int n0 = blockIdx.x * 64;
  const int h = blockIdx.y, b = blockIdx.z;
  const long qbase = ((long)b * CDIM + h * HD) * NPIX;
  const long kvb = (long)(b * NHEADS + h) * HD * HD;
  const long kmb = (long)(b * NHEADS + h) * HD;

#pragma unroll
  for (int i = 0; i < 8; ++i) {             // stage q tile, coalesced in n
    int lin = t + i * 256;
    int n = lin & 63, d = lin >> 6;
    qS[n][d] = qbuf[qbase + (long)d * NPIX + n0 + n];
  }
#pragma unroll
  for (int i = 0; i < 4; ++i) {             // stage kv transposed: [e][d]
    int lin = t + i * 256;
    kvT[lin & 31][lin >> 5] = kvbuf[kvb + lin];
  }
  if (t < 32) kms[t] = kmean[kmb + t];
  __syncthreads();

  if (t < 64) {                             // den[n] = sum_d q[d,n]*kmean[d]
    float s = 0.0f;
#pragma unroll
    for (int d = 0; d < HD; ++d) s += qS[t][d] * kms[d];
    dens[t] = s;
  }
  __syncthreads();

  v8f acc = {};
#pragma unroll
  for (int kk = 0; kk < 32; kk += 4) {
    const int kc = kk + 2 * half;
    v2f a  = lds2(&qS[m_off + lm][kc]);
    v2f bb = lds2(&kvT[e_off + lm][kc]);
    acc = wmma4(a, bb, acc);
  }

#pragma unroll
  for (int r = 0; r < 8; ++r) {
    int n_local = m_off + r + 8 * half;
    int e = e_off + lm;
    otile[e][n_local] = acc[r] / (dens[n_local] + 1e-6f);
  }
  __syncthreads();
#pragma unroll
  for (int i = 0; i < 8; ++i) {             // coalesced 64-wide channel rows
    int lin = t + i * 256;
    int n = lin & 63, e = lin >> 6;
    out[((long)b * CDIM + h * HD + e) * NPIX + n0 + n] = otile[e][n];
  }
}

// ---------------------------------------------------------------------------
// Kernel 4: out += depthwise3x3(x, pe_w) + pe_b  (zero padding 1)
// ---------------------------------------------------------------------------
__global__ __launch_bounds__(256)
void pe_kernel(const float* __restrict__ x, const float* __restrict__ pw,
               const float* __restrict__ pb, float* __restrict__ out) {
  long idx = (long)blockIdx.x * 256 + threadIdx.x;
  int xx = (int)(idx % 56);
  int yy = (int)((idx / 56) % 56);
  int c  = (int)((idx / NPIX) % CDIM);
  long bc = idx / NPIX;                     // b*256 + c
  const float* xc = x + bc * NPIX;
  const float* wc = pw + c * 9;
  float s = pb[c];
#pragma unroll
  for (int dy = -1; dy <= 1; ++dy) {
    int y2 = yy + dy;
    if (y2 < 0 || y2 >= 56) continue;
#pragma unroll
    for (int dx = -1; dx <= 1; ++dx) {
      int x2 = xx + dx;
      if (x2 < 0 || x2 >= 56) continue;
      s += xc[y2 * 56 + x2] * wc[(dy + 1) * 3 + (dx + 1)];
    }
  }
  out[idx] += s;
}

// ---------------------------------------------------------------------------
extern "C" void kernel_launch(void* const* d_in, const int* in_sizes, int n_in,
                              void* d_out, int out_size, void* d_ws, size_t ws_size,
                              hipStream_t stream) {
  (void)in_sizes; (void)n_in; (void)out_size; (void)ws_size;
  const float* x   = (const float*)d_in[0];
  const float* qkw = (const float*)d_in[1];
  const float* qkb = (const float*)d_in[2];
  const float* pew = (const float*)d_in[3];
  const float* peb = (const float*)d_in[4];
  float* out = (float*)d_out;

  float* ws = (float*)d_ws;
  const long QK = 32L * CDIM * NPIX;        // 25,690,112 floats each
  float* qbuf   = ws;
  float* kbuf   = ws + QK;
  float* kvbuf  = ws + 2 * QK;
  float* kmeanb = kvbuf + 32L * NHEADS * HD * HD;

  qk_gemm_kernel<<<dim3(49, 4, 32), 256, 0, stream>>>(x, qkw, qkb, qbuf, kbuf);
  kv_kernel<<<dim3(NHEADS, 32), 128, 0, stream>>>(x, kbuf, kvbuf, kmeanb);
  out_kernel<<<dim3(49, NHEADS, 32), 256, 0, stream>>>(qbuf, kvbuf, kmeanb, out);
  pe_kernel<<<(unsigned)(32L * CDIM * NPIX / 256), 256, 0, stream>>>(x, pew, peb, out);
}